// reversi_model_5265629905181
// MI455X (gfx1250) — compile-verified
//
#include <hip/hip_runtime.h>
#include <math.h>

// ---------------------------------------------------------------------------
// Vector types: WMMA f32 16x16x4 uses A 16x4 (2 VGPR), B 4x16 (2 VGPR),
// C/D 16x16 (8 VGPR). TDM descriptor groups (clang-23 6-arg form):
// (u32x4 g0, i32x8 g1, i32x4 g2, i32x4 g3, i32x8 g4, i32 cpol).
// ---------------------------------------------------------------------------
typedef __attribute__((ext_vector_type(2))) float    v2f;
typedef __attribute__((ext_vector_type(8))) float    v8f;
typedef __attribute__((ext_vector_type(4))) unsigned v4u;
typedef __attribute__((ext_vector_type(8))) int      v8i;
typedef __attribute__((ext_vector_type(4))) int      v4i;

#define SP   80    // S=65 padded to 80 (multiple of 16); pad lanes kept zero
#define H    512
#define KC   64    // K chunk staged in LDS
#define APAD 68    // LDS A row stride (dwords): 64 + 4 pad -> 4*r bank step

#define LDS_A_DW   (64 * APAD)                  // A panel dwords
#define LDS_B_DW   (KC * 16)                    // B panel dwords
#define BUF_DW     (LDS_A_DW + LDS_B_DW)        // one buffer (dwords)
#define LDS_B_OFF  (LDS_A_DW * 4)               // B byte offset inside a buffer
#define LDS_BYTES  (2 * BUF_DW * 4)             // double buffered

// ---------------------------------------------------------------------------
// Issue TDM loads for one K-chunk: A panel (64 dwords x 64 rows, LDS-padded
// to APAD row stride) and B panel (16 dwords x 64 rows, packed).
// ---------------------------------------------------------------------------
__device__ __forceinline__ void tdm_issue_pair(const float* A, const float* B,
                                               int lda, int ldb, int tm, int tn,
                                               int k0, int M, int K,
                                               unsigned ldsBufByte)
{
    v4i z4 = {0, 0, 0, 0};
    v8i z8 = {0, 0, 0, 0, 0, 0, 0, 0};
    {   // ---- A panel ----
        unsigned long long ga = (unsigned long long)(const void*)(A + tm * lda + k0);
        v4u g0;
        g0.x = 1u;                                               // count=1
        g0.y = ldsBufByte;                                       // lds_addr (As)
        g0.z = (unsigned)ga;
        g0.w = (unsigned)((ga >> 32) & 0x01ffffffu) | (2u << 30);// type=2
        unsigned td0 = (unsigned)lda;                            // tensor dim0
        unsigned td1 = (unsigned)(M - tm);                       // rows left; OOB->0
        v8i g1;
        g1[0] = (int)((2u << 16) |                               // data_size=4B
                      (1u << 20) |                               // pad_enable
                      (5u << 22) |                               // pad every 64 dw
                      (3u << 25));                               // pad 4 dw
        g1[1] = (int)((td0 & 0xffffu) << 16);
        g1[2] = (int)((td0 >> 16) | ((td1 & 0xffffu) << 16));
        g1[3] = (int)((td1 >> 16) | (64u << 16));                // tile_dim0 = 64
        g1[4] = 64;                                              // tile_dim1 = 64
        g1[5] = lda;                                             // dim0 stride lo
        g1[6] = 0; g1[7] = 0;
        __builtin_amdgcn_tensor_load_to_lds(g0, g1, z4, z4, z8, 0);
    }
    {   // ---- B panel ----
        unsigned long long gb = (unsigned long long)(const void*)(B + k0 * ldb + tn);
        v4u g0;
        g0.x = 1u;
        g0.y = ldsBufByte + (unsigned)LDS_B_OFF;
        g0.z = (unsigned)gb;
        g0.w = (unsigned)((gb >> 32) & 0x01ffffffu) | (2u << 30);
        unsigned td0 = (unsigned)ldb;
        unsigned td1 = (unsigned)(K - k0);
        v8i g1;
        g1[0] = (int)(2u << 16);                                 // data_size=4B
        g1[1] = (int)((td0 & 0xffffu) << 16);
        g1[2] = (int)((td0 >> 16) | ((td1 & 0xffffu) << 16));
        g1[3] = (int)((td1 >> 16) | (16u << 16));                // tile_dim0 = 16
        g1[4] = 64;                                              // tile_dim1 = 64
        g1[5] = ldb;
        g1[6] = 0; g1[7] = 0;
        __builtin_amdgcn_tensor_load_to_lds(g0, g1, z4, z4, z8, 0);
    }
}

// ---------------------------------------------------------------------------
// Tiled f32 WMMA GEMM: D[M,N] = A(MxK) * B(KxN).
//   TA: A stored [K][M] (A^T given) ; TB: B stored [N][K] (B^T given)
// Workgroup = 4 wave32's = one 64x16 macro-tile.
// NN mode: TDM double-buffered staging (requires K % 64 == 0 — true below),
// so the Tensor Data Mover streams chunk i+1 while WMMA consumes chunk i.
// Transposed modes stage manually (TDM cannot transpose).
// ---------------------------------------------------------------------------
template <bool TA, bool TB>
__global__ void gemm_wmma(const float* __restrict__ A, const float* __restrict__ B,
                          float* __restrict__ D, int M, int N, int K,
                          int lda, int ldb, int ldd)
{
    extern __shared__ float smem[];

    const int lane = threadIdx.x;           // 0..31
    const int wave = threadIdx.y;           // 0..3
    const int tid  = (wave << 5) + lane;    // 0..127

    const int colTiles = N >> 4;
    const int tm = (blockIdx.x / colTiles) << 6;     // macro-tile row base
    const int tn = (blockIdx.x % colTiles) << 4;     // tile col base

    const int half = lane >> 4;              // 0 -> K{0,1}, 1 -> K{2,3}
    const int l16  = lane & 15;
    const int rowW = (wave << 4) + l16;      // lane's A row within macro-tile
    const bool rowValid = (tm + (wave << 4)) < M;    // wave-uniform

    v8f acc = {};

    if (!TA && !TB) {
        // ================= NN: TDM double-buffered pipeline =================
        if (wave == 0)
            tdm_issue_pair(A, B, lda, ldb, tm, tn, 0, M, K, 0u);

        int p = 0;
        for (int k0 = 0; k0 < K; k0 += KC) {
            __builtin_amdgcn_s_wait_tensorcnt(0);   // chunk k0 resident (wave 0)
            __syncthreads();                        // publish to all waves

            if (k0 + KC < K && wave == 0)           // stream next chunk now
                tdm_issue_pair(A, B, lda, ldb, tm, tn, k0 + KC, M, K,
                               (unsigned)((p ^ 1) * (BUF_DW * 4)));

            // keep L2 ahead of the TDM: prefetch the chunk after next
            if (k0 + 2 * KC < K && tid < 64) {
                int row = tm + tid; if (row >= M) row = M - 1;
                __builtin_prefetch(A + row * lda + k0 + 2 * KC, 0, 2);
            }

            if (rowValid) {
                const float* As = smem + p * BUF_DW;
                const float* Bs = As + LDS_A_DW;
                const float* ar = As + rowW * APAD;
#pragma unroll
                for (int kk = 0; kk < KC; kk += 4) {
                    const int ka = kk + (half << 1);
                    v2f a, b;
                    a.x = ar[ka];                    // ds_load b64, conflict-free
                    a.y = ar[ka + 1];
                    b.x = Bs[ka * 16 + l16];         // ds_load 2addr, conflict-free
                    b.y = Bs[(ka + 1) * 16 + l16];
                    acc = __builtin_amdgcn_wmma_f32_16x16x4_f32(
                              false, a, false, b, (short)0, acc, false, false);
                }
            }
            __syncthreads();                        // buf p free for chunk k0+2*KC
            p ^= 1;
        }
    } else {
        // ================= transposed modes: manual staging =================
        float* As = smem;                // [64][APAD]
        float* Bs = smem + LDS_A_DW;     // [KC][16]
        for (int k0 = 0; k0 < K; k0 += KC) {
            const int kc = (K - k0 < KC) ? (K - k0) : KC;
            __syncthreads();

            if (TA) {
                // A stored [K][M]: contiguous along m; kc==64 for all TA calls
                for (int idx = tid; idx < (kc << 6); idx += 128) {
                    int r = idx & 63, k = idx >> 6;
                    int row = tm + r; if (row >= M) row = M - 1;
                    As[r * APAD + k] = A[(k0 + k) * lda + row];
                }
            } else {
                // A row-major: contiguous along k
                const int sh = (kc == KC) ? 6 : 4;
                for (int idx = tid; idx < (kc << 6); idx += 128) {
                    int k = idx & (kc - 1), r = idx >> sh;
                    int row = tm + r; if (row >= M) row = M - 1;
                    As[r * APAD + k] = A[row * lda + k0 + k];
                }
            }
            if (TB) {
                // B stored [N][K]: contiguous along k
                const int sh = (kc == KC) ? 6 : 4;
                for (int idx = tid; idx < (kc << 4); idx += 128) {
                    int k = idx & (kc - 1), n = idx >> sh;
                    Bs[k * 16 + n] = B[(tn + n) * ldb + k0 + k];
                }
            } else {
                // B row-major: contiguous along n
                for (int idx = tid; idx < (kc << 4); idx += 128) {
                    int n = idx & 15, k = idx >> 4;
                    Bs[k * 16 + n] = B[(k0 + k) * ldb + tn + n];
                }
            }
            __syncthreads();

            if (rowValid) {
                const float* ar = As + rowW * APAD;
                if (kc == KC) {
#pragma unroll
                    for (int kk = 0; kk < KC; kk += 4) {
                        const int ka = kk + (half << 1);
                        v2f a, b;
                        a.x = ar[ka];
                        a.y = ar[ka + 1];
                        b.x = Bs[ka * 16 + l16];
                        b.y = Bs[(ka + 1) * 16 + l16];
                        acc = __builtin_amdgcn_wmma_f32_16x16x4_f32(
                                  false, a, false, b, (short)0, acc, false, false);
                    }
                } else {
                    for (int kk = 0; kk < kc; kk += 4) {
                        const int ka = kk + (half << 1);
                        v2f a, b;
                        a.x = ar[ka];
                        a.y = ar[ka + 1];
                        b.x = Bs[ka * 16 + l16];
                        b.y = Bs[(ka + 1) * 16 + l16];
                        acc = __builtin_amdgcn_wmma_f32_16x16x4_f32(
                                  false, a, false, b, (short)0, acc, false, false);
                    }
                }
            }
        }
    }

    if (rowValid) {
        const int dm = tm + (wave << 4) + (half << 3);
#pragma unroll
        for (int r = 0; r < 8; ++r)
            D[(dm + r) * ldd + tn + l16] = acc[r];
    }
}

// ---------------------------------------------------------------------------
// y[h][s] = embedding[h][s] * x[s], x = concat(inputs, 1); pad cols (s>=65)=0
// ---------------------------------------------------------------------------
__global__ void embed_kernel(const float* __restrict__ inp, const float* __restrict__ emb,
                             float* __restrict__ Y)
{
    int idx = blockIdx.x * 256 + threadIdx.x;   // 512*80
    int h = idx / SP, s = idx % SP;
    float x = (s < 64) ? inp[s] : (s == 64 ? 1.f : 0.f);
    float e = (s < 65) ? emb[h * 65 + s] : 0.f;
    Y[idx] = e * x;
}

// Row softmax over j<65 of 80x80 scores; pads written as zero. 1 wave32/row.
__global__ void softmax80(const float* __restrict__ S, float* __restrict__ P)
{
    const int row  = blockIdx.x;
    const int lane = threadIdx.x;
    if (row >= 65) {
        for (int j = lane; j < SP; j += 32) P[row * SP + j] = 0.f;
        return;
    }
    float m = -3.4e38f;
    for (int j = lane; j < 65; j += 32) m = fmaxf(m, S[row * SP + j]);
    for (int o = 16; o > 0; o >>= 1) m = fmaxf(m, __shfl_xor(m, o, 32));
    float sum = 0.f;
    for (int j = lane; j < 65; j += 32) sum += __expf(S[row * SP + j] - m);
    for (int o = 16; o > 0; o >>= 1) sum += __shfl_xor(sum, o, 32);
    const float inv = 1.f / sum;
    for (int j = lane; j < SP; j += 32)
        P[row * SP + j] = (j < 65) ? __expf(S[row * SP + j] - m) * inv : 0.f;
}

// cat[m][k] (80x1536) from three 80x512 attention outputs
__global__ void concat3(const float* __restrict__ a0, const float* __restrict__ a1,
                        const float* __restrict__ a2, float* __restrict__ C)
{
    int idx = blockIdx.x * 256 + threadIdx.x;   // 80*1536
    int k = idx % 1536;
    int m = idx / 1536;
    const float* s = (k < 512) ? a0 : ((k < 1024) ? a1 : a2);
    C[idx] = s[m * 512 + (k & 511)];
}

// Y[h][s] = relu(T[s][h]); pad cols zeroed
__global__ void relu_T(const float* __restrict__ T, float* __restrict__ Y)
{
    int idx = blockIdx.x * 256 + threadIdx.x;   // 512*80
    int h = idx / SP, s = idx % SP;
    float v = (s < 65) ? T[s * 512 + h] : 0.f;
    Y[idx] = fmaxf(v, 0.f);
}

__global__ void relu_row(const float* __restrict__ T64, float* __restrict__ y3)
{
    int j = blockIdx.x * 256 + threadIdx.x;     // 1024
    y3[j] = fmaxf(T64[j], 0.f);
}

// y[n] = tanh(sum_k x[k]*W[k][n]), W: [1024][512]
__global__ void gemv_tanh(const float* __restrict__ x, const float* __restrict__ W,
                          float* __restrict__ y)
{
    int n = blockIdx.x * 256 + threadIdx.x;
    float acc = 0.f;
    for (int k = 0; k < 1024; ++k) acc += x[k] * W[k * 512 + n];
    y[n] = tanhf(acc);
}

// out[n] = sum_k x[k]*W[k][n], W: [512][64]
__global__ void gemv_out(const float* __restrict__ x, const float* __restrict__ W,
                         float* __restrict__ out)
{
    int n = threadIdx.x;
    float acc = 0.f;
    for (int k = 0; k < 512; ++k) acc += x[k] * W[k * 64 + n];
    out[n] = acc;
}

// ---------------------------------------------------------------------------
// Host side
// ---------------------------------------------------------------------------
enum GemmMode { NN, TN, NT };

static void run_gemm(GemmMode mode, const float* A, const float* B, float* D,
                     int M, int N, int K, int lda, int ldb, int ldd, hipStream_t s)
{
    dim3 blk(32, 4);
    dim3 grd(((M + 63) / 64) * (N / 16));
    switch (mode) {
        case NN: gemm_wmma<false, false><<<grd, blk, LDS_BYTES, s>>>(A, B, D, M, N, K, lda, ldb, ldd); break;
        case TN: gemm_wmma<true,  false><<<grd, blk, LDS_BYTES, s>>>(A, B, D, M, N, K, lda, ldb, ldd); break;
        case NT: gemm_wmma<false, true ><<<grd, blk, LDS_BYTES, s>>>(A, B, D, M, N, K, lda, ldb, ldd); break;
    }
}

extern "C" void kernel_launch(void* const* d_in, const int* in_sizes, int n_in,
                              void* d_out, int out_size, void* d_ws, size_t ws_size,
                              hipStream_t stream)
{
    (void)in_sizes; (void)n_in; (void)out_size; (void)ws_size;

    const float* inputs = (const float*)d_in[0];   // [64]
    const float* emb    = (const float*)d_in[1];   // [512,65]
    const float* wq     = (const float*)d_in[2];   // [3,3,512,512]
    const float* wk     = (const float*)d_in[3];
    const float* wv     = (const float*)d_in[4];
    const float* L1     = (const float*)d_in[5];   // [1536,512]
    const float* L2     = (const float*)d_in[6];   // [1536,512]
    const float* L3     = (const float*)d_in[7];   // [1536,1024]
    const float* L4     = (const float*)d_in[8];   // [1024,512]
    const float* L5     = (const float*)d_in[9];   // [512,64]
    float* out = (float*)d_out;

    float* w = (float*)d_ws;
    float* Y   = w;  w += H * SP;     // activations [512][80]
    float* KB  = w;  w += H * SP;
    float* QB  = w;  w += H * SP;
    float* VB  = w;  w += H * SP;
    float* A10 = w;  w += SP * H;     // layer-1 head outputs [80][512]
    float* A11 = w;  w += SP * H;
    float* A12 = w;  w += SP * H;
    float* AX  = w;  w += SP * H;     // layer-2/3 head-2 output (reused)
    float* SC  = w;  w += SP * SP;    // scores [80][80]
    float* SM  = w;  w += SP * SP;    // softmax [80][80]
    float* CAT = w;  w += SP * 1536;  // concat [80][1536]
    float* T   = w;  w += SP * 1024;  // linear output (<=80x1024)
    float* Y3  = w;  w += 1024;
    float* Y4  = w;                   // 512; total ~2.2 MB workspace

    const size_t HH = (size_t)H * H;

    embed_kernel<<<160, 256, 0, stream>>>(inputs, emb, Y);

    auto head = [&](int layer, int h, float* Adst) {
        const float* wki = wk + (size_t)(layer * 3 + h) * HH;
        const float* wqi = wq + (size_t)(layer * 3 + h) * HH;
        const float* wvi = wv + (size_t)(layer * 3 + h) * HH;
        run_gemm(NN, wki, Y, KB, H, SP, H, H, SP, SP, stream);   // K = Wk @ Y
        run_gemm(NN, wqi, Y, QB, H, SP, H, H, SP, SP, stream);   // Q = Wq @ Y
        run_gemm(NN, wvi, Y, VB, H, SP, H, H, SP, SP, stream);   // V = Wv @ Y
        run_gemm(TN, KB, QB, SC, SP, SP, H, SP, SP, SP, stream); // scores = K^T Q
        softmax80<<<SP, 32, 0, stream>>>(SC, SM);
        run_gemm(NT, SM, VB, Adst, SP, H, SP, SP, SP, H, stream);// att = SM V^T
    };

    // Layer 1: all three heads used
    head(0, 0, A10);
    head(0, 1, A11);
    head(0, 2, A12);
    concat3<<<480, 256, 0, stream>>>(A10, A11, A12, CAT);
    run_gemm(NN, CAT, L1, T, SP, 512, 1536, 1536, 512, 512, stream);
    relu_T<<<160, 256, 0, stream>>>(T, Y);

    // Layer 2 (bug preserved: concat a1[0], a1[1], a2[2]) -> only head 2 needed
    head(1, 2, AX);
    concat3<<<480, 256, 0, stream>>>(A10, A11, AX, CAT);
    run_gemm(NN, CAT, L2, T, SP, 512, 1536, 1536, 512, 512, stream);
    relu_T<<<160, 256, 0, stream>>>(T, Y);

    // Layer 3: concat a1[0], a1[1], a3[2]; only row 64 of the linear3 result used
    head(2, 2, AX);
    concat3<<<480, 256, 0, stream>>>(A10, A11, AX, CAT);
    run_gemm(NN, CAT, L3, T, SP, 1024, 1536, 1536, 1024, 1024, stream);
    relu_row<<<4, 256, 0, stream>>>(T + 64 * 1024, Y3);

    gemv_tanh<<<2, 256, 0, stream>>>(Y3, L4, Y4);
    gemv_out<<<1, 64, 0, stream>>>(Y4, L5, out);
}